// UnitaryBranching_54331336294713
// MI455X (gfx1250) — compile-verified
//
#include <hip/hip_runtime.h>

// ---------------------------------------------------------------------------
// UnitaryBranching for MI455X (gfx1250, wave32).
// Matrix engine: V_WMMA_F32_16X16X4_F32 (native fp32 WMMA) for
//   (a) expm via Taylor-Horner + scaling-and-squaring,
//   (b) the per-(token,head) chain products maps = P[pw0] @ P[pw1] @ ...
// ---------------------------------------------------------------------------

#define DIM        128
#define BF         2
#define NHEAD      8
#define MAX_DEPTH  12
#define PAD_DIGIT  3
#define NTOK       256
#define NMATS      (BF * NHEAD + 1)    // 17
#define MELEMS     (DIM * DIM)         // 16384
#define TAYLOR_N   12
#define SQUARINGS  8
#define PRESCALE   0.00390625f         // 2^-8

typedef float v2f __attribute__((ext_vector_type(2)));
typedef float v8f __attribute__((ext_vector_type(8)));

// --------------------------------------------------------------------------
// Path words: root->leaf branch digits of node index, padded with PAD_DIGIT.
// Matches reference _path_words exactly (BF == 2).
// --------------------------------------------------------------------------
__device__ __forceinline__ int make_path(int pos, int* pw) {
  int digs[MAX_DEPTH];
  int d = pos, depth = 0;
#pragma unroll
  for (int t = 0; t < MAX_DEPTH; ++t) {
    if (d > 0) { digs[t] = (d - 1) & (BF - 1); d = (d - 1) >> 1; ++depth; }
    else       { digs[t] = PAD_DIGIT; }
  }
#pragma unroll
  for (int j = 0; j < MAX_DEPTH; ++j)
    pw[j] = (j < depth) ? digs[depth - 1 - j] : PAD_DIGIT;
  return depth;
}

// --------------------------------------------------------------------------
// 128x128x128 fp32 matmul, one 16-wide column strip per wave.
// Wave w owns tiles (i*16, w*16), i = 0..7; accumulators live in VGPRs.
// A-frag (16x4): lane m = L&15, K = k0 + 2*(L>>4) + r   (ISA 7.12.2)
// B-frag (4x16): row K striped across 16 lanes per VGPR (mirrors A halves)
// --------------------------------------------------------------------------
template <typename PA, typename PB>
__device__ __forceinline__ void mm128_tiles(PA A, PB B, v8f acc[8],
                                            int wave, int lane) {
  const int n16  = lane & 15;
  const int half = lane >> 4;
  const int tn   = wave * 16;
#pragma unroll 2
  for (int k0 = 0; k0 < DIM; k0 += 4) {
    const int kk = k0 + 2 * half;
    v2f b;
    b.x = B[kk * DIM + tn + n16];
    b.y = B[(kk + 1) * DIM + tn + n16];
#pragma unroll
    for (int i = 0; i < 8; ++i) {
      v2f a;
      a.x = A[(i * 16 + n16) * DIM + kk];
      a.y = A[(i * 16 + n16) * DIM + kk + 1];
      acc[i] = __builtin_amdgcn_wmma_f32_16x16x4_f32(
          false, a, false, b, (short)0, acc[i], false, false);
    }
  }
}

// C/D layout (ISA 7.12.2): VGPR r -> row r + 8*(lane>>4), col = lane&15.
template <typename PD>
__device__ __forceinline__ void store_tiles(PD D, const v8f acc[8],
                                            int wave, int lane,
                                            float scale, bool addI) {
  const int n16  = lane & 15;
  const int half = lane >> 4;
  const int tn   = wave * 16;
#pragma unroll
  for (int i = 0; i < 8; ++i) {
#pragma unroll
    for (int r = 0; r < 8; ++r) {
      const int row = i * 16 + r + 8 * half;
      const int col = tn + n16;
      float v = acc[i][r] * scale;
      if (addI && row == col) v += 1.0f;
      D[row * DIM + col] = v;
    }
  }
}

// --------------------------------------------------------------------------
// 1) S = (P - P^T) * 2^-SQUARINGS
// --------------------------------------------------------------------------
__global__ void __launch_bounds__(256)
skew_kernel(const float* __restrict__ prim, float* __restrict__ skew) {
  const int mat = blockIdx.x;
  const float* s = prim + mat * MELEMS;
  float* d = skew + mat * MELEMS;
  for (int i = threadIdx.x; i < MELEMS; i += 256) {
    const int r = i >> 7, c = i & (DIM - 1);
    d[i] = (s[i] - s[c * DIM + r]) * PRESCALE;
  }
}

// --------------------------------------------------------------------------
// 2) expm(S) by Horner Taylor (order TAYLOR_N) + SQUARINGS squarings.
//    T lives in LDS; the (L2-resident) skew matrix is the A operand.
//    ||S * 2^-8||_2 ~ 0.036 -> Taylor-12 error ~ eps.
// --------------------------------------------------------------------------
__global__ void __launch_bounds__(256)
expm_kernel(const float* __restrict__ skew, float* __restrict__ prims) {
  __shared__ float Tm[MELEMS];                       // 64 KB
  const int mat  = blockIdx.x;
  const float* B = skew + mat * MELEMS;
  const int tid  = threadIdx.x;
  const int wave = tid >> 5, lane = tid & 31;

  // T = I + B / N
  for (int i = tid; i < MELEMS; i += 256) {
    const int r = i >> 7, c = i & (DIM - 1);
    Tm[i] = B[i] * (1.0f / (float)TAYLOR_N) + ((r == c) ? 1.0f : 0.0f);
  }
  __syncthreads();

  // T = I + (B @ T) / k,  k = N-1 .. 1
  for (int k = TAYLOR_N - 1; k >= 1; --k) {
    v8f acc[8];
#pragma unroll
    for (int i = 0; i < 8; ++i) acc[i] = v8f{};
    mm128_tiles(B, (const float*)Tm, acc, wave, lane);
    __syncthreads();
    store_tiles(Tm, acc, wave, lane, 1.0f / (float)k, true);
    __syncthreads();
  }

  // T = T @ T, repeated
  for (int s = 0; s < SQUARINGS; ++s) {
    v8f acc[8];
#pragma unroll
    for (int i = 0; i < 8; ++i) acc[i] = v8f{};
    mm128_tiles((const float*)Tm, (const float*)Tm, acc, wave, lane);
    __syncthreads();
    store_tiles(Tm, acc, wave, lane, 1.0f, false);
    __syncthreads();
  }

  float* dst = prims + mat * MELEMS;
  for (int i = tid; i < MELEMS; i += 256) dst[i] = Tm[i];
}

// --------------------------------------------------------------------------
// 3) maps[n,h] = P[pw0,h] @ P[pw1,h] @ ... (identity if depth==0).
//    One block per (token, head); M evolves in LDS, next primitive is
//    prefetched into cache (global_prefetch_b8). pw/depth are uniform per
//    block, so EXEC is all-ones around every WMMA. (sos branch of the
//    reference is dead: pw digits are never -1.)
// --------------------------------------------------------------------------
__global__ void __launch_bounds__(256)
maps_kernel(const float* __restrict__ prims, const int* __restrict__ positions,
            float* __restrict__ out) {
  __shared__ float Mm[MELEMS];                       // 64 KB
  const int n = blockIdx.x / NHEAD;
  const int h = blockIdx.x % NHEAD;
  const int tid  = threadIdx.x;
  const int wave = tid >> 5, lane = tid & 31;

  int pw[MAX_DEPTH];
  const int depth = make_path(positions[n], pw);

  if (depth == 0) {
    for (int i = tid; i < MELEMS; i += 256) {
      const int r = i >> 7, c = i & (DIM - 1);
      Mm[i] = (r == c) ? 1.0f : 0.0f;
    }
  } else {
    const float* P0 = prims + (pw[0] * NHEAD + h) * MELEMS;
    for (int i = tid; i < MELEMS; i += 256) Mm[i] = P0[i];
  }
  __syncthreads();

  for (int d = 1; d < depth; ++d) {
    const float* Pd = prims + (pw[d] * NHEAD + h) * MELEMS;
    if (d + 1 < depth) {  // warm next primitive into cache
      const float* Pn = prims + (pw[d + 1] * NHEAD + h) * MELEMS;
      __builtin_prefetch(Pn + (size_t)tid * 64, 0, 1);
    }
    v8f acc[8];
#pragma unroll
    for (int i = 0; i < 8; ++i) acc[i] = v8f{};
    mm128_tiles((const float*)Mm, Pd, acc, wave, lane);   // M = M @ Pd
    __syncthreads();
    store_tiles(Mm, acc, wave, lane, 1.0f, false);
    __syncthreads();
  }

  float* dst = out + (size_t)(n * NHEAD + h) * MELEMS;
  for (int i = tid; i < MELEMS; i += 256) dst[i] = Mm[i];
}

// --------------------------------------------------------------------------
// 4) steps[i,j] = depth_i + depth_j - 2 * common_prefix_len  (as float)
// --------------------------------------------------------------------------
__global__ void __launch_bounds__(256)
steps_kernel(const int* __restrict__ positions, float* __restrict__ steps) {
  const int i = blockIdx.x;
  const int j = threadIdx.x;
  int pwi[MAX_DEPTH], pwj[MAX_DEPTH];
  const int di = make_path(positions[i], pwi);
  const int dj = make_path(positions[j], pwj);
  int cpl = 0;
#pragma unroll
  for (int t = 0; t < MAX_DEPTH; ++t) {
    if (pwi[t] != pwj[t] || pwi[t] == PAD_DIGIT) break;
    ++cpl;
  }
  steps[i * NTOK + j] = (float)(di + dj - 2 * cpl);
}

// --------------------------------------------------------------------------
extern "C" void kernel_launch(void* const* d_in, const int* in_sizes, int n_in,
                              void* d_out, int out_size, void* d_ws,
                              size_t ws_size, hipStream_t stream) {
  const float* prim      = (const float*)d_in[0];   // (17,128,128) f32
  const int*   positions = (const int*)d_in[1];     // (256,) i32
  float*       out       = (float*)d_out;           // maps (33.5M) + steps (65K)

  float* skew  = (float*)d_ws;                      // 17 * 16384 f32
  float* prims = skew + (size_t)NMATS * MELEMS;     // 17 * 16384 f32
  (void)in_sizes; (void)n_in; (void)out_size; (void)ws_size;

  skew_kernel <<<NMATS,        256, 0, stream>>>(prim, skew);
  expm_kernel <<<NMATS,        256, 0, stream>>>(skew, prims);
  maps_kernel <<<NTOK * NHEAD, 256, 0, stream>>>(prims, positions, out);
  steps_kernel<<<NTOK,         256, 0, stream>>>(
      positions, out + (size_t)NTOK * NHEAD * MELEMS);
}